// GCNN_69286412419204
// MI455X (gfx1250) — compile-verified
//
#include <hip/hip_runtime.h>
#include <hip/hip_bf16.h>
#include <math.h>

#define N_NODES 100000
#define N_EDGES 1600000
#define N_GRAPHS 1024
#define D 128

typedef __attribute__((ext_vector_type(2))) float v2f;
typedef __attribute__((ext_vector_type(8))) float v8f;

#define ATOMIC_ADD_F32(p, v) \
    __hip_atomic_fetch_add((p), (v), __ATOMIC_RELAXED, __HIP_MEMORY_SCOPE_AGENT)

// ---------------------------------------------------------------------------
// Degree / normalization
// ---------------------------------------------------------------------------
__global__ void k_init_deg(float* deg, int n) {
    int i = blockIdx.x * blockDim.x + threadIdx.x;
    if (i < n) deg[i] = 1.0f;  // self-loop contributes 1
}

__global__ void k_accum_deg(const int* __restrict__ dst, float* deg, int e) {
    int i = blockIdx.x * blockDim.x + threadIdx.x;
    if (i < e) ATOMIC_ADD_F32(&deg[dst[i]], 1.0f);
}

__global__ void k_rsqrt(const float* __restrict__ deg, float* dis, int n) {
    int i = blockIdx.x * blockDim.x + threadIdx.x;
    if (i < n) dis[i] = rsqrtf(deg[i]);
}

// ---------------------------------------------------------------------------
// WMMA fp32 GEMM:  C[M x 128] = A[M x 128] @ W[128 x 128]
//   MODE 0: val = C[m][n] * dis[m]; out1[m][n] = out2[m][n] = val
//           (out1 = hs for edge gather, out2 = agg initialized with self-loop)
//   MODE 1: out1[m][n] = relu(C[m][n] + bias[n])
// One 16-row M block per 256-thread workgroup; 8 waves cover the 8 N-tiles.
// ---------------------------------------------------------------------------
template <int MODE>
__global__ __launch_bounds__(256) void k_gemm_wmma(
    const float* __restrict__ A, const float* __restrict__ W,
    const float* __restrict__ dis, const float* __restrict__ bias,
    float* __restrict__ out1, float* __restrict__ out2, int M)
{
    const int lane = threadIdx.x & 31;
    const int wave = threadIdx.x >> 5;   // 0..7  -> N tile
    const int m0   = blockIdx.x * 16;
    const int n0   = wave * 16;
    const int half = lane >> 4;          // 0: lanes 0-15, 1: lanes 16-31
    const int r    = lane & 15;
    if (m0 >= M) return;                 // uniform per block, EXEC stays full

    v8f c = {};
    const float* __restrict__ arow = A + (size_t)(m0 + r) * D;

#pragma unroll 4
    for (int k = 0; k < D; k += 4) {
        // A 16x4 f32 fragment: lane r holds A[m0+r][k + 2*half .. +1]
        v2f a = *(const v2f*)(arow + k + 2 * half);
        // B 4x16 f32 fragment: VGPR v holds rows K = v (lanes 0-15) / v+2 (16-31)
        const float* __restrict__ bp = W + (size_t)(k + 2 * half) * D + n0 + r;
        v2f b;
        b.x = bp[0];
        b.y = bp[D];
        c = __builtin_amdgcn_wmma_f32_16x16x4_f32(
                /*neg_a=*/false, a, /*neg_b=*/false, b,
                /*c_mod=*/(short)0, c, /*reuse_a=*/false, /*reuse_b=*/false);
    }

    // C/D layout: VGPR v -> row (v + 8*half), col = lane&15
#pragma unroll
    for (int v = 0; v < 8; ++v) {
        const int row = v + half * 8;
        const size_t idx = (size_t)(m0 + row) * D + n0 + r;
        float val = c[v];
        if (MODE == 0) {
            val *= dis[m0 + row];
            out1[idx] = val;   // hs  (gather source)
            out2[idx] = val;   // agg (self-loop term pre-seeded)
        } else {
            val += bias[n0 + r];
            out1[idx] = val > 0.0f ? val : 0.0f;
        }
    }
}

// ---------------------------------------------------------------------------
// Edge scatter: agg[dst] += hs[src]   (one wave per edge, float4 per lane)
// ---------------------------------------------------------------------------
__global__ __launch_bounds__(256) void k_edge_scatter(
    const int* __restrict__ src, const int* __restrict__ dst,
    const float* __restrict__ hs, float* __restrict__ agg, int E)
{
    const int e    = blockIdx.x * 8 + (threadIdx.x >> 5);
    const int lane = threadIdx.x & 31;
    if (e >= E) return;
    const int s = src[e];
    const int d = dst[e];
    const float4 v = *(const float4*)(hs + (size_t)s * D + lane * 4);
    float* p = agg + (size_t)d * D + lane * 4;
    ATOMIC_ADD_F32(p + 0, v.x);
    ATOMIC_ADD_F32(p + 1, v.y);
    ATOMIC_ADD_F32(p + 2, v.z);
    ATOMIC_ADD_F32(p + 3, v.w);
}

// ---------------------------------------------------------------------------
// Layer epilogue: h = relu(dis[node] * agg + bias)
// ---------------------------------------------------------------------------
__global__ void k_finalize(const float* __restrict__ agg,
                           const float* __restrict__ dis,
                           const float* __restrict__ bias,
                           float* __restrict__ h, int n)
{
    int idx = blockIdx.x * blockDim.x + threadIdx.x;
    if (idx < n) {
        const int i = idx >> 7;
        const int c = idx & (D - 1);
        float v = dis[i] * agg[idx] + bias[c];
        h[idx] = v > 0.0f ? v : 0.0f;
    }
}

// ---------------------------------------------------------------------------
// Mean pool over batch segments
// ---------------------------------------------------------------------------
__global__ void k_zero(float* p, int n) {
    int i = blockIdx.x * blockDim.x + threadIdx.x;
    if (i < n) p[i] = 0.0f;
}

__global__ void k_pool_accum(const float* __restrict__ h,
                             const int* __restrict__ batch,
                             float* __restrict__ psum, float* __restrict__ pcnt)
{
    const int i = blockIdx.x;      // node
    const int c = threadIdx.x;     // feature
    const int g = batch[i];
    ATOMIC_ADD_F32(&psum[(size_t)g * D + c], h[(size_t)i * D + c]);
    if (c == 0) ATOMIC_ADD_F32(&pcnt[g], 1.0f);
}

__global__ void k_pool_div(const float* __restrict__ psum,
                           const float* __restrict__ pcnt,
                           float* __restrict__ pooled, int n)
{
    int idx = blockIdx.x * blockDim.x + threadIdx.x;
    if (idx < n) pooled[idx] = psum[idx] / fmaxf(pcnt[idx >> 7], 1.0f);
}

// ---------------------------------------------------------------------------
// fc2 (128 -> 2) + log_softmax
// ---------------------------------------------------------------------------
__global__ void k_head(const float* __restrict__ g, const float* __restrict__ w2,
                       const float* __restrict__ b2, float* __restrict__ out)
{
    int gi = blockIdx.x * blockDim.x + threadIdx.x;
    if (gi >= N_GRAPHS) return;
    float l0 = b2[0], l1 = b2[1];
    const float* row = g + (size_t)gi * D;
    for (int k = 0; k < D; ++k) {
        const float v = row[k];
        l0 += v * w2[k * 2 + 0];
        l1 += v * w2[k * 2 + 1];
    }
    const float m = fmaxf(l0, l1);
    const float lse = m + logf(expf(l0 - m) + expf(l1 - m));
    out[gi * 2 + 0] = l0 - lse;
    out[gi * 2 + 1] = l1 - lse;
}

// ---------------------------------------------------------------------------
extern "C" void kernel_launch(void* const* d_in, const int* in_sizes, int n_in,
                              void* d_out, int out_size, void* d_ws, size_t ws_size,
                              hipStream_t stream)
{
    const float* x      = (const float*)d_in[0];
    const int*   ei     = (const int*)d_in[1];   // [2, E]
    const int*   batch  = (const int*)d_in[2];
    const float* W1     = (const float*)d_in[3];
    const float* b1     = (const float*)d_in[4];
    const float* W2     = (const float*)d_in[5];
    const float* b2     = (const float*)d_in[6];
    const float* fc1_w  = (const float*)d_in[7];
    const float* fc1_b  = (const float*)d_in[8];
    const float* fc2_w  = (const float*)d_in[9];
    const float* fc2_b  = (const float*)d_in[10];
    float*       out    = (float*)d_out;

    const int* esrc = ei;
    const int* edst = ei + N_EDGES;

    // Workspace partition (floats)
    float* ws     = (float*)d_ws;
    float* deg    = ws;                              // N
    float* dis    = deg  + N_NODES;                  // N
    float* hs     = dis  + N_NODES;                  // N*D
    float* agg    = hs   + (size_t)N_NODES * D;      // N*D
    float* h      = agg  + (size_t)N_NODES * D;      // N*D
    float* psum   = h    + (size_t)N_NODES * D;      // G*D
    float* pcnt   = psum + (size_t)N_GRAPHS * D;     // G
    float* pooled = pcnt + N_GRAPHS;                 // G*D
    float* gout   = pooled + (size_t)N_GRAPHS * D;   // G*D

    const int NB  = (N_NODES + 255) / 256;
    const int EB  = (N_EDGES + 255) / 256;
    const int NDB = (N_NODES * D + 255) / 256;

    // Degree + normalization
    k_init_deg<<<NB, 256, 0, stream>>>(deg, N_NODES);
    k_accum_deg<<<EB, 256, 0, stream>>>(edst, deg, N_EDGES);
    k_rsqrt<<<NB, 256, 0, stream>>>(deg, dis, N_NODES);

    // Layer 1: hs = (x@W1)*dis, agg = hs (self-loop); agg += hs[src]; relu(dis*agg + b1)
    k_gemm_wmma<0><<<N_NODES / 16, 256, 0, stream>>>(x, W1, dis, nullptr, hs, agg, N_NODES);
    k_edge_scatter<<<(N_EDGES + 7) / 8, 256, 0, stream>>>(esrc, edst, hs, agg, N_EDGES);
    k_finalize<<<NDB, 256, 0, stream>>>(agg, dis, b1, h, N_NODES * D);

    // Layer 2
    k_gemm_wmma<0><<<N_NODES / 16, 256, 0, stream>>>(h, W2, dis, nullptr, hs, agg, N_NODES);
    k_edge_scatter<<<(N_EDGES + 7) / 8, 256, 0, stream>>>(esrc, edst, hs, agg, N_EDGES);
    k_finalize<<<NDB, 256, 0, stream>>>(agg, dis, b2, h, N_NODES * D);

    // Mean pool
    k_zero<<<(N_GRAPHS * D + N_GRAPHS + 255) / 256, 256, 0, stream>>>(psum, N_GRAPHS * D + N_GRAPHS);
    k_pool_accum<<<N_NODES, D, 0, stream>>>(h, batch, psum, pcnt);
    k_pool_div<<<(N_GRAPHS * D + 255) / 256, 256, 0, stream>>>(psum, pcnt, pooled, N_GRAPHS * D);

    // fc1 (WMMA) + fc2 + log_softmax
    k_gemm_wmma<1><<<N_GRAPHS / 16, 256, 0, stream>>>(pooled, fc1_w, nullptr, fc1_b, gout, nullptr, N_GRAPHS);
    k_head<<<(N_GRAPHS + 255) / 256, 256, 0, stream>>>(gout, fc2_w, fc2_b, out);
}